// CR8_reg_2_stage_76682346102797
// MI455X (gfx1250) — compile-verified
//
#include <hip/hip_runtime.h>

// ---------------------------------------------------------------------------
// Sizes fixed by the reference
// ---------------------------------------------------------------------------
#define NTOK 49152           // 192*256 pixels / tokens
#define HH   192
#define WW   256
#define C0   16              // coarse classes
#define C1   16              // fine classes
#define TSTEP 64                          // tokens per block-tile (4 waves x 16)
#define NTB64_MAX (NTOK / TSTEP + C0)       // 784 worst-case stage-B block tiles
#define NTC64_MAX (NTOK / TSTEP + C0 * C1)  // 1024 worst-case stage-C block tiles

typedef __attribute__((ext_vector_type(2))) float v2f;
typedef __attribute__((ext_vector_type(8))) float v8f;
typedef unsigned int u32;
typedef __attribute__((ext_vector_type(4))) u32 v4u32;
typedef __attribute__((ext_vector_type(4))) int v4i32;
typedef __attribute__((ext_vector_type(8))) int v8i32;

__device__ __forceinline__ float lrelu(float v) { return v > 0.0f ? v : 0.01f * v; }

// ---------------------------------------------------------------------------
// TDM staging (see CDNA5 ISA): DMA nelem contiguous
// floats into LDS. Compile-time diagnostics below report which path was built.
// ---------------------------------------------------------------------------
#if defined(__has_builtin)
#if __has_builtin(__builtin_amdgcn_tensor_load_to_lds)
#define HAVE_TDM 1
#endif
#endif
#ifndef HAVE_TDM
#define HAVE_TDM 0
#endif

#define STR2(x) #x
#define STR(x) STR2(x)
#if HAVE_TDM
#pragma message("CDNA5-DIAG: TDM builtin path ENABLED, clang=" STR(__clang_major__))
#else
#pragma message("CDNA5-DIAG: TDM builtin path DISABLED (fallback copy), clang=" STR(__clang_major__))
#endif

__device__ __forceinline__ void stage_to_lds(float* ldst, u32 lds_byte_off,
                                             const float* gsrc, int nelem) {
#if HAVE_TDM
  const unsigned long long ga = (unsigned long long)gsrc;
  v4u32 g0;
  g0[0] = (u32)__builtin_amdgcn_readfirstlane(1);                 // count=1 (valid)
  g0[1] = (u32)__builtin_amdgcn_readfirstlane((int)lds_byte_off); // lds_addr
  g0[2] = (u32)__builtin_amdgcn_readfirstlane((int)(ga & 0xffffffffull));
  g0[3] = (u32)__builtin_amdgcn_readfirstlane((int)((u32)(ga >> 32) | (2u << 30))); // type=2
  v8i32 g1;
  g1[0] = __builtin_amdgcn_readfirstlane((int)(2u << 16));           // data_size=4B
  g1[1] = __builtin_amdgcn_readfirstlane((int)((u32)nelem << 16));   // tensor_dim0[15:0]
  g1[2] = __builtin_amdgcn_readfirstlane((int)(1u << 16));           // tensor_dim1=1
  g1[3] = __builtin_amdgcn_readfirstlane((int)((u32)nelem << 16));   // tile_dim0
  g1[4] = __builtin_amdgcn_readfirstlane(1);                         // tile_dim1=1
  g1[5] = __builtin_amdgcn_readfirstlane(nelem);                     // dim0_stride
  g1[6] = 0;
  g1[7] = 0;
  v4i32 z4 = {0, 0, 0, 0};
#if __clang_major__ >= 23
  v8i32 z8 = {0, 0, 0, 0, 0, 0, 0, 0};
  __builtin_amdgcn_tensor_load_to_lds(g0, g1, z4, z4, z8, 0);
#else
  __builtin_amdgcn_tensor_load_to_lds(g0, g1, z4, z4, 0);
#endif
  (void)ldst;
#else
  for (int i = (int)threadIdx.x; i < nelem; i += (int)blockDim.x)
    ldst[i] = gsrc[i];
#endif
}

__device__ __forceinline__ void stage_wait_sync() {
#if HAVE_TDM
  __builtin_amdgcn_s_wait_tensorcnt(0);
#endif
  __syncthreads();
}

// ---------------------------------------------------------------------------
// Dense stage: per-pixel 128->128->128->17 via fp32 WMMA (16x16x4, chained K).
// Two 16-pixel tiles per wave: each weight A-fragment load feeds two WMMAs,
// halving weight traffic (the inner loop was weight-load paced).
// ---------------------------------------------------------------------------
__device__ __forceinline__ v8f wmma4(v2f a, v2f b, v8f c) {
  return __builtin_amdgcn_wmma_f32_16x16x4_f32(false, a, false, b, (short)0, c,
                                               false, false);
}

__device__ __forceinline__ void dense_layer2x(
    const float* __restrict__ W, const float* __restrict__ B,
    const v2f* bf0, const v2f* bf1, float* act0, float* act1,
    int hi, int lx)
{
#pragma unroll 1
  for (int mt = 0; mt < 8; ++mt) {
    const float* wrow = W + (mt * 16 + lx) * 128 + hi * 2;
    v8f acc0, acc1;
#pragma unroll
    for (int g = 0; g < 8; ++g) {
      const float b = B[mt * 16 + hi * 8 + g];
      acc0[g] = b; acc1[g] = b;
    }
#pragma unroll
    for (int kc = 0; kc < 32; ++kc) {
      v2f a = *(const v2f*)(wrow + kc * 4);
      acc0 = wmma4(a, bf0[kc], acc0);
      acc1 = wmma4(a, bf1[kc], acc1);
    }
#pragma unroll
    for (int g = 0; g < 8; ++g) {
      act0[(mt * 16 + hi * 8 + g) * 16 + lx] = lrelu(acc0[g]);
      act1[(mt * 16 + hi * 8 + g) * 16 + lx] = lrelu(acc1[g]);
    }
  }
}

__global__ __launch_bounds__(64) void dense_wmma_kernel(
    const float* __restrict__ x_in,
    const float* __restrict__ w11, const float* __restrict__ b11,
    const float* __restrict__ w12, const float* __restrict__ b12,
    const float* __restrict__ w13, const float* __restrict__ b13,
    float* __restrict__ out_mask, int* __restrict__ inds1)
{
  __shared__ float sAct[2][2][128 * 16];   // [wave][tile] 32 KB

  const int lane = threadIdx.x & 31;
  const int wid  = threadIdx.x >> 5;       // 0..1
  const int p0   = (blockIdx.x * 2 + wid) * 32;   // two 16-px tiles
  if (p0 >= NTOK) return;

  const int hi = lane >> 4;
  const int lx = lane & 15;
  float* act0 = sAct[wid][0];
  float* act1 = sAct[wid][1];

  // ---- Layer 1: B fragments from global (coalesced) for both tiles ----
  v2f bf0[32], bf1[32];
#pragma unroll
  for (int kc = 0; kc < 32; ++kc) {
    const int kb = kc * 4 + hi * 2;
    bf0[kc].x = x_in[(kb + 0) * NTOK + p0 + lx];
    bf0[kc].y = x_in[(kb + 1) * NTOK + p0 + lx];
    bf1[kc].x = x_in[(kb + 0) * NTOK + p0 + 16 + lx];
    bf1[kc].y = x_in[(kb + 1) * NTOK + p0 + 16 + lx];
  }
  dense_layer2x(w11, b11, bf0, bf1, act0, act1, hi, lx);

  // ---- Layer 2 ----
#pragma unroll
  for (int kc = 0; kc < 32; ++kc) {
    const int kb = kc * 4 + hi * 2;
    bf0[kc].x = act0[(kb + 0) * 16 + lx];
    bf0[kc].y = act0[(kb + 1) * 16 + lx];
    bf1[kc].x = act1[(kb + 0) * 16 + lx];
    bf1[kc].y = act1[(kb + 1) * 16 + lx];
  }
  dense_layer2x(w12, b12, bf0, bf1, act0, act1, hi, lx);

  // ---- Layer 3: logits + mask ----
#pragma unroll
  for (int kc = 0; kc < 32; ++kc) {
    const int kb = kc * 4 + hi * 2;
    bf0[kc].x = act0[(kb + 0) * 16 + lx];
    bf0[kc].y = act0[(kb + 1) * 16 + lx];
    bf1[kc].x = act1[(kb + 0) * 16 + lx];
    bf1[kc].y = act1[(kb + 1) * 16 + lx];
  }
  v8f accL0, accL1;
#pragma unroll
  for (int g = 0; g < 8; ++g) { accL0[g] = b13[hi * 8 + g]; accL1[g] = accL0[g]; }
  {
    const float* wrow = w13 + lx * 128 + hi * 2;
#pragma unroll
    for (int kc = 0; kc < 32; ++kc) {
      v2f a = *(const v2f*)(wrow + kc * 4);
      accL0 = wmma4(a, bf0[kc], accL0);
      accL1 = wmma4(a, bf1[kc], accL1);
    }
  }
  // per-pixel argmax (first-max), tile 0 then tile 1
  float bv0 = accL0[0]; int bi0 = hi * 8;
  float bv1 = accL1[0]; int bi1 = hi * 8;
#pragma unroll
  for (int g = 1; g < 8; ++g) {
    const int i = hi * 8 + g;
    if (accL0[g] > bv0) { bv0 = accL0[g]; bi0 = i; }
    if (accL1[g] > bv1) { bv1 = accL1[g]; bi1 = i; }
  }
  {
    const float ov = __shfl_xor(bv0, 16, 32);
    const int   oi = __shfl_xor(bi0, 16, 32);
    if (ov > bv0 || (ov == bv0 && oi < bi0)) { bv0 = ov; bi0 = oi; }
  }
  {
    const float ov = __shfl_xor(bv1, 16, 32);
    const int   oi = __shfl_xor(bi1, 16, 32);
    if (ov > bv1 || (ov == bv1 && oi < bi1)) { bv1 = ov; bi1 = oi; }
  }

  v8f accM0, accM1;
#pragma unroll
  for (int g = 0; g < 8; ++g) { accM0[g] = b13[16]; accM1[g] = b13[16]; }
  {
    const float* wrow = w13 + 16 * 128 + hi * 2;
#pragma unroll
    for (int kc = 0; kc < 32; ++kc) {
      v2f a = *(const v2f*)(wrow + kc * 4);
      accM0 = wmma4(a, bf0[kc], accM0);
      accM1 = wmma4(a, bf1[kc], accM1);
    }
  }
  if (lane < 16) {
    inds1[p0 + lane]          = bi0;
    inds1[p0 + 16 + lane]     = bi1;
    out_mask[p0 + lane]       = lrelu(accM0[0]);
    out_mask[p0 + 16 + lane]  = lrelu(accM1[0]);
  }
}

// ---------------------------------------------------------------------------
// Bucketing helpers
// ---------------------------------------------------------------------------
__global__ void zero_kernel(int* __restrict__ p, int n) {
  const int i = blockIdx.x * blockDim.x + threadIdx.x;
  if (i < n) p[i] = 0;
}

__global__ __launch_bounds__(256) void hist_kernel(const int* __restrict__ cls,
                                                   int* __restrict__ cnt) {
  const int n = blockIdx.x * 256 + threadIdx.x;
  if (n < NTOK) atomicAdd(&cnt[cls[n]], 1);
}

__global__ void scan_kernel(const int* __restrict__ cnt, int* __restrict__ off,
                            int* __restrict__ tile, int* __restrict__ nTiles,
                            int nc, int step) {
  if (blockIdx.x != 0 || threadIdx.x != 0) return;
  int o = 0;
  for (int c = 0; c < nc; ++c) { off[c] = o; o += cnt[c]; }
  off[nc] = o;
  int t = 0;
  for (int c = 0; c < nc; ++c) {
    const int cn = cnt[c];
    for (int j = 0; j < cn; j += step) {
      tile[2 * t]     = c;
      tile[2 * t + 1] = off[c] + j;
      ++t;
    }
  }
  *nTiles = t;
}

__global__ __launch_bounds__(256) void scatter_kernel(
    const int* __restrict__ cls, const int* __restrict__ off,
    int* __restrict__ cur, int* __restrict__ order) {
  const int n = blockIdx.x * 256 + threadIdx.x;
  if (n < NTOK) {
    const int c = cls[n];
    const int p = atomicAdd(&cur[c], 1);
    order[off[c] + p] = n;
  }
}

// ---------------------------------------------------------------------------
// Stage B: one block (4 waves, 64 tokens) per class tile.
// Expert weights TDM-staged to LDS once per block; MLP 128->32->32->16 + argmax.
// ---------------------------------------------------------------------------
__global__ __launch_bounds__(128) void stageB_wmma_kernel(
    const float* __restrict__ x_in,
    const float* __restrict__ wc21, const float* __restrict__ bc21,
    const float* __restrict__ wc22, const float* __restrict__ bc22,
    const float* __restrict__ wc23, const float* __restrict__ bc23,
    const int* __restrict__ order, const int* __restrict__ off,
    const int* __restrict__ tile, const int* __restrict__ nTiles,
    int* __restrict__ inds12)
{
  __shared__ float smem[4096 + 1024 + 512 + 4 * 512];   // w1 | w2 | w3 | act
  const int t = blockIdx.x;
  if (t >= *nTiles) return;

  float* sw1 = smem;            // LDS byte offset 0
  float* sw2 = smem + 4096;     // 16384
  float* sw3 = smem + 5120;     // 20480
  const int lane = threadIdx.x & 31;
  const int wid  = threadIdx.x >> 5;
  float* act = smem + 5632 + wid * 512;
  const int hi = lane >> 4;
  const int lx = lane & 15;

  const int cls    = tile[2 * t];
  const int start0 = tile[2 * t + 1];
  const int end    = off[cls + 1];

  // kick off weight DMA, overlap with token gathers
  stage_to_lds(sw1, 0u,     wc21 + cls * 4096, 4096);   // [128,32]
  stage_to_lds(sw2, 16384u, wc22 + cls * 1024, 1024);   // [32,32]
  stage_to_lds(sw3, 20480u, wc23 + cls * 512,  512);    // [32,16]

  const int start = start0 + wid * 16;
  const int idx   = start + lx;
  const int cidx  = idx < end ? idx : end - 1;          // pad: duplicate last
  const int tokn  = order[cidx];
  const int perm  = (tokn % HH) * WW + (tokn / HH);     // faithful transpose(1,3)

  v2f bf[32];
#pragma unroll
  for (int kc = 0; kc < 32; ++kc) {
    const int kb = kc * 4 + hi * 2;
    bf[kc].x = x_in[(kb + 0) * NTOK + perm];
    bf[kc].y = x_in[(kb + 1) * NTOK + perm];
  }
  stage_wait_sync();

  // ---- L1: 128 -> 32 ----
#pragma unroll
  for (int mt = 0; mt < 2; ++mt) {
    const float* wbase = sw1 + mt * 16 + lx;
    v8f acc;
#pragma unroll
    for (int g = 0; g < 8; ++g) acc[g] = bc21[cls * 32 + mt * 16 + hi * 8 + g];
#pragma unroll
    for (int kc = 0; kc < 32; ++kc) {
      const int kb = kc * 4 + hi * 2;
      v2f a; a.x = wbase[kb * 32]; a.y = wbase[(kb + 1) * 32];
      acc = wmma4(a, bf[kc], acc);
    }
#pragma unroll
    for (int g = 0; g < 8; ++g)
      act[(mt * 16 + hi * 8 + g) * 16 + lx] = lrelu(acc[g]);
  }

  // ---- L2: 32 -> 32 ----
  v2f bf2[8];
#pragma unroll
  for (int kc = 0; kc < 8; ++kc) {
    const int kb = kc * 4 + hi * 2;
    bf2[kc].x = act[(kb + 0) * 16 + lx];
    bf2[kc].y = act[(kb + 1) * 16 + lx];
  }
#pragma unroll
  for (int mt = 0; mt < 2; ++mt) {
    const float* wbase = sw2 + mt * 16 + lx;
    v8f acc;
#pragma unroll
    for (int g = 0; g < 8; ++g) acc[g] = bc22[cls * 32 + mt * 16 + hi * 8 + g];
#pragma unroll
    for (int kc = 0; kc < 8; ++kc) {
      const int kb = kc * 4 + hi * 2;
      v2f a; a.x = wbase[kb * 32]; a.y = wbase[(kb + 1) * 32];
      acc = wmma4(a, bf2[kc], acc);
    }
#pragma unroll
    for (int g = 0; g < 8; ++g)
      act[(mt * 16 + hi * 8 + g) * 16 + lx] = lrelu(acc[g]);
  }

  // ---- L3: 32 -> 16 logits ----
#pragma unroll
  for (int kc = 0; kc < 8; ++kc) {
    const int kb = kc * 4 + hi * 2;
    bf2[kc].x = act[(kb + 0) * 16 + lx];
    bf2[kc].y = act[(kb + 1) * 16 + lx];
  }
  v8f accL;
#pragma unroll
  for (int g = 0; g < 8; ++g) accL[g] = bc23[cls * 16 + hi * 8 + g];
  {
    const float* wbase = sw3 + lx;
#pragma unroll
    for (int kc = 0; kc < 8; ++kc) {
      const int kb = kc * 4 + hi * 2;
      v2f a; a.x = wbase[kb * 16]; a.y = wbase[(kb + 1) * 16];
      accL = wmma4(a, bf2[kc], accL);
    }
  }

  float bestv = accL[0];
  int   besti = hi * 8;
#pragma unroll
  for (int g = 1; g < 8; ++g) {
    const float v = accL[g];
    const int   i = hi * 8 + g;
    if (v > bestv) { bestv = v; besti = i; }
  }
  const float ov = __shfl_xor(bestv, 16, 32);
  const int   oi = __shfl_xor(besti, 16, 32);
  if (ov > bestv || (ov == bestv && oi < besti)) { bestv = ov; besti = oi; }

  if (lane < 16 && (start + lane) < end)
    inds12[order[start + lane]] = cls * C0 + besti;
}

// ---------------------------------------------------------------------------
// Stage C: one block (4 waves, 64 tokens) per inds12 tile.
// Expert weights TDM-staged; MLP 128->32->16->1 (last dot also WMMA).
// ---------------------------------------------------------------------------
__global__ __launch_bounds__(128) void stageC_wmma_kernel(
    const float* __restrict__ x_in,
    const float* __restrict__ wr11, const float* __restrict__ br11,
    const float* __restrict__ wr12, const float* __restrict__ br12,
    const float* __restrict__ wr13, const float* __restrict__ br13,
    const int* __restrict__ order, const int* __restrict__ off,
    const int* __restrict__ tile, const int* __restrict__ nTiles,
    float* __restrict__ out_real)
{
  __shared__ float smem[4096 + 512 + 4 * 512];          // w1 | w2 | act
  const int t = blockIdx.x;
  if (t >= *nTiles) return;

  float* sw1 = smem;            // LDS byte offset 0
  float* sw2 = smem + 4096;     // 16384
  const int lane = threadIdx.x & 31;
  const int wid  = threadIdx.x >> 5;
  float* act = smem + 4608 + wid * 512;
  const int hi = lane >> 4;
  const int lx = lane & 15;

  const int cls    = tile[2 * t];                       // 0..255
  const int start0 = tile[2 * t + 1];
  const int end    = off[cls + 1];

  stage_to_lds(sw1, 0u,     wr11 + cls * 4096, 4096);   // [128,32]
  stage_to_lds(sw2, 16384u, wr12 + cls * 512,  512);    // [32,16]

  const int start = start0 + wid * 16;
  const int idx   = start + lx;
  const int cidx  = idx < end ? idx : end - 1;
  const int tokn  = order[cidx];
  const int perm  = (tokn % HH) * WW + (tokn / HH);

  v2f bf[32];
#pragma unroll
  for (int kc = 0; kc < 32; ++kc) {
    const int kb = kc * 4 + hi * 2;
    bf[kc].x = x_in[(kb + 0) * NTOK + perm];
    bf[kc].y = x_in[(kb + 1) * NTOK + perm];
  }
  stage_wait_sync();

  // ---- L1: 128 -> 32 ----
#pragma unroll
  for (int mt = 0; mt < 2; ++mt) {
    const float* wbase = sw1 + mt * 16 + lx;
    v8f acc;
#pragma unroll
    for (int g = 0; g < 8; ++g) acc[g] = br11[cls * 32 + mt * 16 + hi * 8 + g];
#pragma unroll
    for (int kc = 0; kc < 32; ++kc) {
      const int kb = kc * 4 + hi * 2;
      v2f a; a.x = wbase[kb * 32]; a.y = wbase[(kb + 1) * 32];
      acc = wmma4(a, bf[kc], acc);
    }
#pragma unroll
    for (int g = 0; g < 8; ++g)
      act[(mt * 16 + hi * 8 + g) * 16 + lx] = lrelu(acc[g]);
  }

  // ---- L2: 32 -> 16 ----
  v2f bf2[8];
#pragma unroll
  for (int kc = 0; kc < 8; ++kc) {
    const int kb = kc * 4 + hi * 2;
    bf2[kc].x = act[(kb + 0) * 16 + lx];
    bf2[kc].y = act[(kb + 1) * 16 + lx];
  }
  {
    const float* wbase = sw2 + lx;
    v8f acc;
#pragma unroll
    for (int g = 0; g < 8; ++g) acc[g] = br12[cls * 16 + hi * 8 + g];
#pragma unroll
    for (int kc = 0; kc < 8; ++kc) {
      const int kb = kc * 4 + hi * 2;
      v2f a; a.x = wbase[kb * 16]; a.y = wbase[(kb + 1) * 16];
      acc = wmma4(a, bf2[kc], acc);
    }
#pragma unroll
    for (int g = 0; g < 8; ++g)
      act[(hi * 8 + g) * 16 + lx] = lrelu(acc[g]);
  }

  // ---- L3: 16 -> 1 (all A-rows read the single weight row) ----
  v2f bf3[4];
#pragma unroll
  for (int kc = 0; kc < 4; ++kc) {
    const int kb = kc * 4 + hi * 2;
    bf3[kc].x = act[(kb + 0) * 16 + lx];
    bf3[kc].y = act[(kb + 1) * 16 + lx];
  }
  v8f accR;
#pragma unroll
  for (int g = 0; g < 8; ++g) accR[g] = br13[cls];
  {
    const float* wrow = wr13 + cls * 16;
#pragma unroll
    for (int kc = 0; kc < 4; ++kc) {
      const int kb = kc * 4 + hi * 2;
      v2f a; a.x = wrow[kb]; a.y = wrow[kb + 1];
      accR = wmma4(a, bf3[kc], accR);
    }
  }

  if (lane < 16 && (start + lane) < end) {
    const int n = order[start + lane];
    out_real[n] = ((float)cls + accR[0]) * (1.0f / (float)(C0 * C1));
  }
}

// ---------------------------------------------------------------------------
// Launcher
// ---------------------------------------------------------------------------
extern "C" void kernel_launch(void* const* d_in, const int* in_sizes, int n_in,
                              void* d_out, int out_size, void* d_ws, size_t ws_size,
                              hipStream_t stream) {
  const float* x_in = (const float*)d_in[0];
  const float* w11  = (const float*)d_in[1];
  const float* b11  = (const float*)d_in[2];
  const float* w12  = (const float*)d_in[3];
  const float* b12  = (const float*)d_in[4];
  const float* w13  = (const float*)d_in[5];
  const float* b13  = (const float*)d_in[6];
  const float* wc21 = (const float*)d_in[7];
  const float* bc21 = (const float*)d_in[8];
  const float* wc22 = (const float*)d_in[9];
  const float* bc22 = (const float*)d_in[10];
  const float* wc23 = (const float*)d_in[11];
  const float* bc23 = (const float*)d_in[12];
  const float* wr11 = (const float*)d_in[13];
  const float* br11 = (const float*)d_in[14];
  const float* wr12 = (const float*)d_in[15];
  const float* br12 = (const float*)d_in[16];
  const float* wr13 = (const float*)d_in[17];
  const float* br13 = (const float*)d_in[18];

  float* out_real = (float*)d_out;
  float* out_mask = out_real + NTOK;

  // workspace layout (ints), ~804 KB total
  int* wsI    = (int*)d_ws;
  int* inds1  = wsI;
  int* inds12 = wsI + NTOK;
  int* orderB = wsI + 2 * NTOK;
  int* orderC = wsI + 3 * NTOK;
  int* cnt16  = wsI + 4 * NTOK;          // counters block: 16+16+256+256 = 544
  int* cur16  = cnt16 + 16;
  int* cnt256 = cur16 + 16;
  int* cur256 = cnt256 + 256;
  int* off16  = cur256 + 256;            // 17
  int* off256 = off16 + 17;              // 257
  int* tileB  = off256 + 257;            // 2*NTB64_MAX
  int* nTB    = tileB + 2 * NTB64_MAX;   // 1
  int* tileC  = nTB + 1;                 // 2*NTC64_MAX
  int* nTC    = tileC + 2 * NTC64_MAX;   // 1

  // 1) dense stage -> mask, inds1   (2 waves x 2 tiles = 64 px per block)
  dense_wmma_kernel<<<NTOK / 64, 64, 0, stream>>>(
      x_in, w11, b11, w12, b12, w13, b13, out_mask, inds1);

  // 2) reset all counters
  zero_kernel<<<3, 256, 0, stream>>>(cnt16, 544);

  // 3) bucket by inds1 (16 classes), 64-token block tiles
  hist_kernel<<<NTOK / 256, 256, 0, stream>>>(inds1, cnt16);
  scan_kernel<<<1, 1, 0, stream>>>(cnt16, off16, tileB, nTB, C0, TSTEP);
  scatter_kernel<<<NTOK / 256, 256, 0, stream>>>(inds1, off16, cur16, orderB);

  // 4) stage B (bucketed WMMA + TDM-staged weights) -> inds12
  stageB_wmma_kernel<<<NTB64_MAX, 128, 0, stream>>>(
      x_in, wc21, bc21, wc22, bc22, wc23, bc23,
      orderB, off16, tileB, nTB, inds12);

  // 5) bucket by inds12 (256 classes), 64-token block tiles
  hist_kernel<<<NTOK / 256, 256, 0, stream>>>(inds12, cnt256);
  scan_kernel<<<1, 1, 0, stream>>>(cnt256, off256, tileC, nTC, C0 * C1, TSTEP);
  scatter_kernel<<<NTOK / 256, 256, 0, stream>>>(inds12, off256, cur256, orderC);

  // 6) stage C (bucketed WMMA + TDM-staged weights) -> x_real
  stageC_wmma_kernel<<<NTC64_MAX, 128, 0, stream>>>(
      x_in, wr11, br11, wr12, br12, wr13, br13,
      orderC, off256, tileC, nTC, out_real);
}